// Net_86689619902948
// MI455X (gfx1250) — compile-verified
//
#include <hip/hip_runtime.h>

typedef float v2f __attribute__((ext_vector_type(2)));
typedef float v8f __attribute__((ext_vector_type(8)));
typedef int v4i __attribute__((ext_vector_type(4)));

#define BN_EPS 1e-5f

#if __has_builtin(__builtin_amdgcn_global_load_async_to_lds_b128)
#define HAS_ASYNC_LDS 1
#else
#define HAS_ASYNC_LDS 0
#endif

typedef __attribute__((address_space(1))) v4i* gv4i_ptr;  // global v4i*
typedef __attribute__((address_space(3))) v4i* lv4i_ptr;  // LDS v4i*

// D = A(16x4) * B(4x16) + C, fp32 matrix core op (V_WMMA_F32_16X16X4_F32)
__device__ __forceinline__ v8f wmma_f32(v2f a, v2f b, v8f c) {
  return __builtin_amdgcn_wmma_f32_16x16x4_f32(false, a, false, b, (short)0, c,
                                               false, false);
}

__device__ __forceinline__ void wait_async() {
#if HAS_ASYNC_LDS
#if __has_builtin(__builtin_amdgcn_s_wait_asynccnt)
  __builtin_amdgcn_s_wait_asynccnt(0);
#else
  asm volatile("s_wait_asynccnt 0x0" ::: "memory");
#endif
#endif
}

// Stage one 16B chunk global->LDS (async DMA path on gfx1250), or write zeros
// for out-of-bounds halo chunks (disjoint addresses -> no ordering hazard).
__device__ __forceinline__ void stage_chunk(const float* __restrict__ gsrc,
                                            float* __restrict__ ldst,
                                            bool ok) {
  if (ok) {
#if HAS_ASYNC_LDS
    v4i* gs = (v4i*)gsrc;  // generic typed, then explicit addrspace cast
    v4i* ls = (v4i*)ldst;
    __builtin_amdgcn_global_load_async_to_lds_b128((gv4i_ptr)gs, (lv4i_ptr)ls,
                                                   0, 0);
#else
    *(float4*)ldst = *(const float4*)gsrc;
#endif
  } else {
    float4 z = {0.0f, 0.0f, 0.0f, 0.0f};
    *(float4*)ldst = z;
  }
}

// ================= conv1 (3->16, K=27 pad 28) + bn1 + relu + 2x2 pool =====
// x: [16,3,512,512] -> h: [16,16,256,256]
// Block: 256 conv-x of one pooled row. LDS tile: 3ch x 4rows x 264 (halo=0).
// Wave: 32 conv-x (2 M-tiles) x 2 rows = 4 accumulators = 28 WMMAs.

template <int k, int j>
__device__ __forceinline__ int aoff1(int hlf) {  // LDS byte offset, const-sel
  constexpr int Kp0 = 4 * k + j;
  constexpr int Kp1 = 4 * k + 2 + j;
  constexpr int K0 = (Kp0 < 27) ? Kp0 : 0;
  constexpr int K1 = (Kp1 < 27) ? Kp1 : 0;
  constexpr int C0 = (K0 / 9) * 1056 + ((K0 % 9) / 3) * 264 + (K0 % 9) % 3;
  constexpr int C1 = (K1 / 9) * 1056 + ((K1 % 9) / 3) * 264 + (K1 % 9) % 3;
  return (hlf ? C1 : C0) << 2;
}

template <int k>
__device__ __forceinline__ void conv1_step(const char* aBase,
                                           const float* __restrict__ w,
                                           int hlf, int m, v8f& c00, v8f& c01,
                                           v8f& c10, v8f& c11) {
  const int K0 = 4 * k + 2 * hlf;
  v2f bf;  // B element (K', n=lane&15) = conv1_w[n*27 + K']
  bf.x = w[m * 27 + K0];
  if constexpr (4 * k + 3 < 27) {
    bf.y = w[m * 27 + K0 + 1];
  } else {  // k==6, hlf==1 -> K'=27 pad: zero the weight (A*0 == 0)
    const float wy = w[m * 27 + (hlf ? K0 : K0 + 1)];
    bf.y = hlf ? 0.0f : wy;
  }
  const float* p0 = (const float*)(aBase + aoff1<k, 0>(hlf));
  const float* p1 = (const float*)(aBase + aoff1<k, 1>(hlf));
  // rows: +264 words; M-tile1: +16 words -> pure ds immediate offsets
  v2f a00 = {p0[0],   p1[0]};    // row0, Mtile0
  v2f a01 = {p0[16],  p1[16]};   // row0, Mtile1
  v2f a10 = {p0[264], p1[264]};  // row1, Mtile0
  v2f a11 = {p0[280], p1[280]};  // row1, Mtile1
  c00 = wmma_f32(a00, bf, c00);
  c01 = wmma_f32(a01, bf, c01);
  c10 = wmma_f32(a10, bf, c10);
  c11 = wmma_f32(a11, bf, c11);
}

__global__ __launch_bounds__(256) void conv1_bn_relu_pool(
    const float* __restrict__ x, const float* __restrict__ w,
    const float* __restrict__ bg, const float* __restrict__ bb,
    const float* __restrict__ bm, const float* __restrict__ bv,
    float* __restrict__ hout) {
  __shared__ float tA[3 * 1056];  // [ch][4 rows][264], 12.7 KB

  const int tid = threadIdx.x;
  const int bi = blockIdx.x;
  const int xh = bi & 1;
  const int py = (bi >> 1) & 255;
  const int b  = bi >> 9;
  const int X0 = xh << 8;
  const int y0 = py << 1;
  const float* xb = x + b * (3 << 18);

  // stage input tile: 12 (ch,row) pairs x 66 chunks of 4 floats
  for (int i = tid; i < 792; i += 256) {
    const int pair = i / 66;
    const int seg  = i - pair * 66;
    const int ch = pair >> 2;
    const int r3 = pair & 3;
    const int r  = y0 - 1 + r3;
    const int gx0 = X0 - 4 + (seg << 2);
    const bool ok = ((unsigned)r < 512u) & ((unsigned)gx0 < 512u);
    const int gofs = (ch << 18) + (r << 9) + gx0;
    stage_chunk(xb + (ok ? gofs : 0), tA + ch * 1056 + r3 * 264 + (seg << 2),
                ok);
  }
  wait_async();
  __syncthreads();

  const int wv = tid >> 5;
  const int lane = tid & 31;
  const int hlf = lane >> 4;
  const int m   = lane & 15;
  const int xloc = (wv << 5) + m;  // local conv-x of M-tile0
  const char* aBase = (const char*)tA + ((xloc + 3) << 2);

  v8f c00 = {0, 0, 0, 0, 0, 0, 0, 0};
  v8f c01 = c00, c10 = c00, c11 = c00;

  conv1_step<0>(aBase, w, hlf, m, c00, c01, c10, c11);
  conv1_step<1>(aBase, w, hlf, m, c00, c01, c10, c11);
  conv1_step<2>(aBase, w, hlf, m, c00, c01, c10, c11);
  conv1_step<3>(aBase, w, hlf, m, c00, c01, c10, c11);
  conv1_step<4>(aBase, w, hlf, m, c00, c01, c10, c11);
  conv1_step<5>(aBase, w, hlf, m, c00, c01, c10, c11);
  conv1_step<6>(aBase, w, hlf, m, c00, c01, c10, c11);

  // bn (per channel = lane&15) -> 2x2 maxpool in registers -> relu -> store
  const int ch = m;
  const float inv = bg[ch] * rsqrtf(bv[ch] + BN_EPS);
  const float sh  = bb[ch] - bm[ch] * inv;
#pragma unroll
  for (int mt = 0; mt < 2; ++mt) {
    v8f r0 = mt ? c01 : c00;
    v8f r1 = mt ? c11 : c10;
    float4 o;
    float* po = &o.x;
#pragma unroll
    for (int j = 0; j < 4; ++j) {
      float e00 = fmaf(inv, r0[2 * j], sh);
      float e01 = fmaf(inv, r0[2 * j + 1], sh);
      float e10 = fmaf(inv, r1[2 * j], sh);
      float e11 = fmaf(inv, r1[2 * j + 1], sh);
      po[j] = fmaxf(fmaxf(fmaxf(e00, e01), fmaxf(e10, e11)), 0.0f);
    }
    const int px = (X0 >> 1) + (wv << 4) + (mt << 3) + (hlf << 2);
    *(float4*)&hout[((b * 16 + ch) << 16) + (py << 8) + px] = o;
  }
}

// ================= conv2 (16->32, K=144) + bn2 + relu =====================
// h: [16,16,256,256] -> feat: [16,32,256,256]
// Block: 128 x positions x 2 rows; 8 waves: wave = 32 x (2 M-tiles) x 1 row
// x 32 ch (2 N-tiles) = 4 accumulators = 144 WMMAs.
// LDS: tile 16ch x 4rows x 136 (34 KB) + weights stride-148 (18.5 KB).

template <int k, int j>
__device__ __forceinline__ int aoff2(int hlf) {
  constexpr int Kp0 = 4 * k + j;
  constexpr int Kp1 = 4 * k + 2 + j;
  constexpr int C0 = (Kp0 / 9) * 544 + ((Kp0 % 9) / 3) * 136 + (Kp0 % 9) % 3;
  constexpr int C1 = (Kp1 / 9) * 544 + ((Kp1 % 9) / 3) * 136 + (Kp1 % 9) % 3;
  return (hlf ? C1 : C0) << 2;
}

template <int k>
__device__ __forceinline__ void conv2_steps(const char* aBase,
                                            const float* __restrict__ wp,
                                            int hlf, v8f& c00, v8f& c01,
                                            v8f& c10, v8f& c11) {
  if constexpr (k < 36) {
    const float* p0 = (const float*)(aBase + aoff2<k, 0>(hlf));
    const float* p1 = (const float*)(aBase + aoff2<k, 1>(hlf));
    v2f a0 = {p0[0],  p1[0]};   // M-tile0
    v2f a1 = {p0[16], p1[16]};  // M-tile1 (+64B imm)
    // weights: stride 148 (conflict-free), pure immediate offsets
    const float2 w0 = *(const float2*)&wp[4 * k];         // oc = m
    const float2 w1 = *(const float2*)&wp[2368 + 4 * k];  // oc = m+16
    v2f b0 = {w0.x, w0.y};
    v2f b1 = {w1.x, w1.y};
    c00 = wmma_f32(a0, b0, c00);
    c01 = wmma_f32(a1, b0, c01);
    c10 = wmma_f32(a0, b1, c10);
    c11 = wmma_f32(a1, b1, c11);
    conv2_steps<k + 1>(aBase, wp, hlf, c00, c01, c10, c11);
  }
}

__global__ __launch_bounds__(256) void conv2_bn_relu(
    const float* __restrict__ h, const float* __restrict__ w,
    const float* __restrict__ bg, const float* __restrict__ bb,
    const float* __restrict__ bm, const float* __restrict__ bv,
    float* __restrict__ feat) {
  __shared__ float tA[16 * 544];  // [ch][4 rows][136]
  __shared__ float wl[32 * 148];  // padded weight rows

  const int tid = threadIdx.x;
  const int bi = blockIdx.x;
  const int xh = bi & 1;
  const int yp = (bi >> 1) & 127;
  const int b  = bi >> 8;
  const int X0 = xh << 7;
  const float* hb = h + (b << 20);

  // stage weights: 32 oc x 36 chunks, repacked to stride 148
  for (int i = tid; i < 1152; i += 256) {
    const int oc = i / 36;
    const int cs = i - oc * 36;
    stage_chunk(w + oc * 144 + (cs << 2), wl + oc * 148 + (cs << 2), true);
  }
  // stage input tile: 64 (ch,row) pairs x 34 chunks
  for (int i = tid; i < 2176; i += 256) {
    const int pair = i / 34;
    const int seg  = i - pair * 34;
    const int ch = pair >> 2;
    const int r3 = pair & 3;
    const int r  = (yp << 1) - 1 + r3;
    const int gx0 = X0 - 4 + (seg << 2);
    const bool ok = ((unsigned)r < 256u) & ((unsigned)gx0 < 256u);
    const int gofs = (ch << 16) + (r << 8) + gx0;
    stage_chunk(hb + (ok ? gofs : 0), tA + ch * 544 + r3 * 136 + (seg << 2),
                ok);
  }
  wait_async();
  __syncthreads();

  const int wv = tid >> 5;
  const int lane = tid & 31;
  const int hlf = lane >> 4;
  const int m   = lane & 15;
  const int trow = wv >> 2;               // which of the 2 output rows
  const int y    = (yp << 1) + trow;
  const int xloc = ((wv & 3) << 5) + m;   // local x of M-tile0
  const char* aBase = (const char*)tA + ((xloc + 3 + trow * 136) << 2);
  const float* wp = wl + m * 148 + 2 * hlf;

  v8f c00 = {0, 0, 0, 0, 0, 0, 0, 0};
  v8f c01 = c00, c10 = c00, c11 = c00;

  conv2_steps<0>(aBase, wp, hlf, c00, c01, c10, c11);

#pragma unroll
  for (int nt = 0; nt < 2; ++nt) {
    const int ch = m + (nt << 4);
    const float inv = bg[ch] * rsqrtf(bv[ch] + BN_EPS);
    const float sh  = bb[ch] - bm[ch] * inv;
#pragma unroll
    for (int mt = 0; mt < 2; ++mt) {
      v8f cc = nt ? (mt ? c11 : c10) : (mt ? c01 : c00);
      float4 o0, o1;
      float* p0 = &o0.x;
      float* p1 = &o1.x;
#pragma unroll
      for (int j = 0; j < 4; ++j) {
        p0[j] = fmaxf(fmaf(inv, cc[j],     sh), 0.0f);
        p1[j] = fmaxf(fmaf(inv, cc[j + 4], sh), 0.0f);
      }
      const int xg = X0 + ((wv & 3) << 5) + (mt << 4) + (hlf << 3);
      float* dst = &feat[((b * 32 + ch) << 16) + (y << 8) + xg];
      *(float4*)dst       = o0;
      *(float4*)(dst + 4) = o1;
    }
  }
}

// ================= proposal mask from feat[0] (float4 vectorized) =========
__global__ __launch_bounds__(256) void pf_mask_kernel(
    const float* __restrict__ feat, const float* __restrict__ pfw,
    const float* __restrict__ pfb, float* __restrict__ mask,
    float* __restrict__ mpart) {
  __shared__ float red[256];
  const int t = threadIdx.x;
  const int hw = (blockIdx.x * 256 + t) << 2;
  const float bias = pfb[0];
  float4 sc = {bias, bias, bias, bias};
#pragma unroll
  for (int c = 0; c < 32; ++c) {
    const float4 f = *(const float4*)&feat[(c << 16) + hw];
    const float wv = pfw[c];
    sc.x = fmaf(f.x, wv, sc.x);
    sc.y = fmaf(f.y, wv, sc.y);
    sc.z = fmaf(f.z, wv, sc.z);
    sc.w = fmaf(f.w, wv, sc.w);
  }
  float4 mv;  // sigmoid(s) > 0.5  <=>  s > 0
  mv.x = sc.x > 0.0f ? 1.0f : 0.0f;
  mv.y = sc.y > 0.0f ? 1.0f : 0.0f;
  mv.z = sc.z > 0.0f ? 1.0f : 0.0f;
  mv.w = sc.w > 0.0f ? 1.0f : 0.0f;
  *(float4*)&mask[hw] = mv;
  red[t] = mv.x + mv.y + mv.z + mv.w;
  __syncthreads();
  for (int s = 128; s > 0; s >>= 1) {
    if (t < s) red[t] += red[t + s];
    __syncthreads();
  }
  if (t == 0) mpart[blockIdx.x] = red[0];
}

// ================= masked channel sums (float4 vectorized) ================
__global__ __launch_bounds__(256) void masked_sum_kernel(
    const float* __restrict__ feat, const float* __restrict__ mask,
    float* __restrict__ spart) {
  __shared__ float red[256];
  const int t = threadIdx.x;
  const int chunk = blockIdx.x & 15;
  const int c = (blockIdx.x >> 4) & 31;
  const int b = blockIdx.x >> 9;
  const int hw0 = chunk << 12;
  const float4* fp = (const float4*)(feat + ((b * 32 + c) << 16) + hw0);
  const float4* mp = (const float4*)(mask + hw0);
  float acc = 0.0f;
#pragma unroll
  for (int i = 0; i < 4; ++i) {
    const int idx = (i << 8) + t;
    const float4 f = fp[idx];
    const float4 mm = mp[idx];
    acc = fmaf(f.x, mm.x, acc);
    acc = fmaf(f.y, mm.y, acc);
    acc = fmaf(f.z, mm.z, acc);
    acc = fmaf(f.w, mm.w, acc);
  }
  red[t] = acc;
  __syncthreads();
  for (int st = 128; st > 0; st >>= 1) {
    if (t < st) red[t] += red[t + st];
    __syncthreads();
  }
  if (t == 0) spart[blockIdx.x] = red[0];
}

// ================= deterministic final reduction ==========================
__global__ __launch_bounds__(1024) void reduce_kernel(
    const float* __restrict__ spart, const float* __restrict__ mpart,
    float* __restrict__ s, float* __restrict__ msum) {
  const int i = threadIdx.x;
  if (i < 512) {
    float a = 0.0f;
#pragma unroll
    for (int k = 0; k < 16; ++k) a += spart[i * 16 + k];
    s[i] = a;
  } else if (i == 512) {
    float a = 0.0f;
    for (int k = 0; k < 64; ++k) a += mpart[k];
    msum[0] = a;
  }
}

// ================= head: vocab (masked mean) + classifier =================
__global__ __launch_bounds__(128) void head_kernel(
    const float* __restrict__ s, const float* __restrict__ msum,
    const float* __restrict__ vw, const float* __restrict__ vb,
    const float* __restrict__ cw, const float* __restrict__ cb,
    float* __restrict__ out) {
  __shared__ float ss[32];
  __shared__ float cm[80];
  const int b = blockIdx.x;
  const int t = threadIdx.x;
  if (t < 32) ss[t] = s[b * 32 + t];
  __syncthreads();
  const float ms  = msum[0];
  const float cnt = fmaxf(ms, 1.0f);
  if (t < 80) {
    float a = 0.0f;
#pragma unroll
    for (int c = 0; c < 32; ++c) a = fmaf(vw[t * 32 + c], ss[c], a);
    cm[t] = a / cnt + vb[t] * (ms / cnt);
  }
  __syncthreads();
  if (t < 80) {
    float o = cb[t];
    for (int k = 0; k < 80; ++k) o = fmaf(cm[k], cw[t * 80 + k], o);
    out[b * 80 + t] = o;
  }
}

extern "C" void kernel_launch(void* const* d_in, const int* in_sizes, int n_in,
                              void* d_out, int out_size, void* d_ws,
                              size_t ws_size, hipStream_t stream) {
  (void)in_sizes; (void)n_in; (void)out_size; (void)ws_size;
  const float* x    = (const float*)d_in[0];
  const float* c1w  = (const float*)d_in[1];
  const float* bn1g = (const float*)d_in[2];
  const float* bn1b = (const float*)d_in[3];
  const float* bn1m = (const float*)d_in[4];
  const float* bn1v = (const float*)d_in[5];
  const float* c2w  = (const float*)d_in[6];
  const float* bn2g = (const float*)d_in[7];
  const float* bn2b = (const float*)d_in[8];
  const float* bn2m = (const float*)d_in[9];
  const float* bn2v = (const float*)d_in[10];
  const float* pfw  = (const float*)d_in[11];
  const float* pfb  = (const float*)d_in[12];
  const float* vw   = (const float*)d_in[13];
  const float* vb   = (const float*)d_in[14];
  const float* cw   = (const float*)d_in[15];
  const float* cb   = (const float*)d_in[16];

  float* ws    = (float*)d_ws;
  float* h     = ws;                    // 16*16*256*256 = 16,777,216
  float* feat  = h + 16777216;          // 16*32*256*256 = 33,554,432
  float* mask  = feat + 33554432;       // 65,536
  float* s     = mask + 65536;          // 512
  float* msum  = s + 512;               // 1
  float* mpart = msum + 1;              // 64
  float* spart = mpart + 64;            // 8192

  // conv1: 16 b x 256 pooled rows x 2 halves = 8192 blocks
  conv1_bn_relu_pool<<<8192, 256, 0, stream>>>(x, c1w, bn1g, bn1b, bn1m, bn1v,
                                               h);
  // conv2: 16 b x 128 row-pairs x 2 halves = 4096 blocks
  conv2_bn_relu<<<4096, 256, 0, stream>>>(h, c2w, bn2g, bn2b, bn2m, bn2v,
                                          feat);
  pf_mask_kernel<<<64, 256, 0, stream>>>(feat, pfw, pfb, mask, mpart);
  masked_sum_kernel<<<8192, 256, 0, stream>>>(feat, mask, spart);
  reduce_kernel<<<1, 1024, 0, stream>>>(spart, mpart, s, msum);
  head_kernel<<<16, 128, 0, stream>>>(s, msum, vw, vb, cw, cb, (float*)d_out);
}